// ChamferLoss_69526930588393
// MI455X (gfx1250) — compile-verified
//
#include <hip/hip_runtime.h>
#include <math.h>

typedef __attribute__((ext_vector_type(2))) float v2f;
typedef __attribute__((ext_vector_type(8))) float v8f;

#define NPTS 32768
#define TILE 256                      // B points staged in LDS per outer iteration
#define WAVES_PER_WG 8
#define ROWS_PER_WG (WAVES_PER_WG * 16)   // 128 rows per workgroup

// For each row i of A, compute min_j ||A_i - B_j||^2 entirely inside
// V_WMMA_F32_16X16X4_F32 via homogeneous 4D operands:
//   A_row = (ax, ay, az, 1), B_col = (-2bx, -2by, -2bz, ||b||^2), C = ||a||^2
//   => D = ||a||^2 - 2 a.b + ||b||^2 = squared distance.
__global__ __launch_bounds__(256) void chamfer_min_kernel(
    const float* __restrict__ A, const float* __restrict__ B,
    float* __restrict__ dmin, int n, int m)
{
    // One 16-byte record per staged B point: {-2bx, -2by, -2bz, b2}
    __shared__ __align__(16) float lb[TILE * 4];

    const int tid  = threadIdx.x;
    const int wave = tid >> 5;
    const int lane = tid & 31;
    const int half = (lane >= 16) ? 1 : 0;   // which K-half / M-half this lane holds
    const int l16  = lane & 15;

    const int rowBase = blockIdx.x * ROWS_PER_WG + wave * 16;
    const int myRow   = rowBase + l16;

    // ---- A operand (16x4 fp32): lanes 0-15 -> K=0,1 = (ax,ay); lanes 16-31 -> K=2,3 = (az,1)
    const float ax = A[3 * myRow + 0];
    const float ay = A[3 * myRow + 1];
    const float az = A[3 * myRow + 2];
    const float naL = ax * ax + ay * ay + az * az;   // ||A_row||^2 for row rowBase + lane%16

    v2f avec;
    avec.x = half ? az : ax;
    avec.y = half ? 1.0f : ay;

    // C initializer: C VGPR r holds row (r + 8*half) -> broadcast that row's norm.
    v8f cinit;
#pragma unroll
    for (int r = 0; r < 8; ++r)
        cinit[r] = __shfl(naL, r + half * 8, 32);

    float best[8];
#pragma unroll
    for (int r = 0; r < 8; ++r) best[r] = 3.0e38f;

    // Preload first tile's B point into registers.
    float pbx, pby, pbz;
    {
        const int j = tid;
        pbx = B[3 * j + 0];
        pby = B[3 * j + 1];
        pbz = B[3 * j + 2];
    }

    for (int base = 0; base < m; base += TILE) {
        // Stage transformed record for this tile.
        float4 w;
        w.x = -2.0f * pbx;
        w.y = -2.0f * pby;
        w.z = -2.0f * pbz;
        w.w = pbx * pbx + pby * pby + pbz * pbz;
        *(float4*)&lb[tid * 4] = w;
        __syncthreads();

        // Issue next tile's global loads now; they complete under the WMMA work.
        if (base + TILE < m) {
            const int j = base + TILE + tid;
            pbx = B[3 * j + 0];
            pby = B[3 * j + 1];
            pbz = B[3 * j + 2];
        }

        // 16 column blocks; every LDS access is an immediate-offset ds_load_b64,
        // conflict-free across all 64 banks.
#pragma unroll
        for (int cb = 0; cb < TILE / 16; ++cb) {
            const v2f bvec = *(const v2f*)&lb[cb * 64 + l16 * 4 + half * 2];
            const v8f d = __builtin_amdgcn_wmma_f32_16x16x4_f32(
                /*neg_a=*/false, avec, /*neg_b=*/false, bvec,
                /*c_mod=*/(short)0, cinit, /*reuse_a=*/false, /*reuse_b=*/false);
#pragma unroll
            for (int r = 0; r < 8; ++r)
                best[r] = fminf(best[r], d[r]);
        }
        __syncthreads();
    }

    // min across the 16 lanes of each half-wave (columns of the C block)
#pragma unroll
    for (int r = 0; r < 8; ++r) {
        float v = best[r];
#pragma unroll
        for (int off = 8; off >= 1; off >>= 1)
            v = fminf(v, __shfl_xor(v, off, 32));
        best[r] = v;
    }
    if (l16 == 0) {
#pragma unroll
        for (int r = 0; r < 8; ++r)
            dmin[rowBase + half * 8 + r] = fmaxf(best[r], 0.0f); // clamp fp cancellation
    }
}

// sum(sqrt(d1))/n, sum(sqrt(d2))/n, combine with decay scaling -> scalar loss
__global__ __launch_bounds__(256) void chamfer_reduce_kernel(
    const float* __restrict__ d1, const float* __restrict__ d2,
    const int* __restrict__ curp, float* __restrict__ out, int n)
{
    __shared__ float s1s[256], s2s[256];
    float s1 = 0.0f, s2 = 0.0f;
    for (int i = threadIdx.x; i < n; i += 256) {
        s1 += sqrtf(d1[i]);
        s2 += sqrtf(d2[i]);
    }
    s1s[threadIdx.x] = s1;
    s2s[threadIdx.x] = s2;
    __syncthreads();
    for (int off = 128; off > 0; off >>= 1) {
        if (threadIdx.x < off) {
            s1s[threadIdx.x] += s1s[threadIdx.x + off];
            s2s[threadIdx.x] += s2s[threadIdx.x + off];
        }
        __syncthreads();
    }
    if (threadIdx.x == 0) {
        const int cur = curp[0];
        const float scale = 10.0f / powf(1.02f, (float)(cur / 20));
        out[0] = (s1s[0] / (float)n + s2s[0] / (float)n) * 0.5f * scale;
    }
}

extern "C" void kernel_launch(void* const* d_in, const int* in_sizes, int n_in,
                              void* d_out, int out_size, void* d_ws, size_t ws_size,
                              hipStream_t stream) {
    const float* target = (const float*)d_in[0];  // [32768,3] fp32
    const float* output = (const float*)d_in[1];  // [32768,3] fp32
    const int*   cur    = (const int*)d_in[2];    // scalar
    float* out = (float*)d_out;

    float* dist1 = (float*)d_ws;          // [NPTS]
    float* dist2 = dist1 + NPTS;          // [NPTS]

    dim3 grid(NPTS / ROWS_PER_WG);        // 256 workgroups
    chamfer_min_kernel<<<grid, 256, 0, stream>>>(target, output, dist1, NPTS, NPTS);
    chamfer_min_kernel<<<grid, 256, 0, stream>>>(output, target, dist2, NPTS, NPTS);
    chamfer_reduce_kernel<<<1, 256, 0, stream>>>(dist1, dist2, cur, out, NPTS);
}